// SNNLayer_72430328479969
// MI455X (gfx1250) — compile-verified
//
#include <hip/hip_runtime.h>

// ---------------------------------------------------------------------------
// SNN surrogate-gradient scan: m/s recurrence over T=64 steps, elementwise in
// (B,N)=(32,16384). Pure bandwidth-bound streaming scan (~402MB, ~17us floor
// at 23.3 TB/s). No matmul content -> WMMA inapplicable; CDNA5 path used is
// the async global->LDS copy engine (ASYNCcnt, 4-deep ring prefetch).
// ---------------------------------------------------------------------------

typedef __attribute__((ext_vector_type(4))) float f4;
typedef __attribute__((ext_vector_type(4))) int   v4i;

#define T_STEPS 64
#define NPIX    16384
#define BN      (32 * 16384)            // B*N = 524288 elements per timestep
#define VEC     4
#define BLOCK   256
#define NBLOCKS ((BN / VEC) / BLOCK)    // 512 blocks
#define DEPTH   4                       // async prefetch ring depth (steps)

#define DECAY   0.9f
#define THRESH  0.5f
#define RSCALE  0.1f
#define SLOPE   8.0f
#define LOG2E   1.4426950408889634f

#if __has_builtin(__builtin_amdgcn_global_load_async_to_lds_b128) && \
    __has_builtin(__builtin_amdgcn_s_wait_asynccnt)
#define HAVE_ASYNC_LDS 1
#else
#define HAVE_ASYNC_LDS 0
#warning "gfx1250 async-to-LDS builtins unavailable; using register double-buffer fallback"
#endif

// Builtin signature: (v4i* global AS1, v4i* LDS AS3, imm offset, imm cpol).
typedef v4i __attribute__((address_space(1)))* g_v4i_p;   // global (AS1)
typedef v4i __attribute__((address_space(3)))* l_v4i_p;   // LDS (AS3)

// One recurrence step on a float4 lane-bundle.
//   m' = 0.9*m + g*r*w + s*c          (c = 0.1*rw - 0.5 precomputed)
//   s' = 1 / (1 + exp2((4 - 8*m') * log2(e)))   -> v_exp_f32 + v_rcp_f32
__device__ __forceinline__ f4 snn_step(f4& m, f4 s, f4 g, f4 r, f4 w, f4 c) {
  m = DECAY * m + g * r * w + s * c;
  f4 xn = (SLOPE * THRESH * LOG2E) - (SLOPE * LOG2E) * m;  // = -8*(m-0.5)*log2e
  f4 sn;
#pragma unroll
  for (int i = 0; i < 4; ++i) {
    float e = __builtin_amdgcn_exp2f(xn[i]);     // v_exp_f32
    sn[i] = __builtin_amdgcn_rcpf(1.0f + e);     // v_rcp_f32
  }
  return sn;
}

extern "C" __global__ __launch_bounds__(BLOCK)
void snn_scan_kernel(const float* __restrict__ m0,
                     const float* __restrict__ s0,
                     const float* __restrict__ gate,
                     const float* __restrict__ gamma,
                     const float* __restrict__ mw,
                     const float* __restrict__ rw,
                     float* __restrict__ out) {
  const int tid = threadIdx.x;
  const int idx = blockIdx.x * BLOCK + tid;   // float4 slot in [0, BN/4)
  const int e0  = idx * VEC;                  // element index in [0, BN)
  const int n0  = e0 & (NPIX - 1);            // pixel index (VEC divides NPIX)

  // Per-n constants, hoisted out of the T loop.
  const f4 w = *(const f4*)(mw + n0);
  const f4 c = RSCALE * (*(const f4*)(rw + n0)) - THRESH;

  // Carried state in VGPRs.
  f4 m = *(const f4*)(m0 + e0);
  f4 s = *(const f4*)(s0 + e0);

#if HAVE_ASYNC_LDS
  // DEPTH-deep async global->LDS ring: each step's gate/gamma DMA is issued
  // DEPTH-1 iterations before consumption, so per-wave HBM latency tolerance
  // is ~3 iterations of VALU work on top of multi-wave occupancy. Each lane
  // reads back only the 16B it requested -> per-wave s_wait_asynccnt is the
  // only synchronization needed (copies complete in issue order).
  __shared__ f4 sg[DEPTH][BLOCK];   // gate slots
  __shared__ f4 sr[DEPTH][BLOCK];   // gamma slots (4*2*256*16B = 32 KB/block)

  auto issue = [&](int t, int slot) {
    // Global side: integer round-trip (global AS is identity-mapped in the
    // generic space). LDS side: C-style addrspacecast generic -> AS3.
    __builtin_amdgcn_global_load_async_to_lds_b128(
        (g_v4i_p)(size_t)(gate + (size_t)t * BN + e0),
        (l_v4i_p)&sg[slot][tid], 0, 0);
    __builtin_amdgcn_global_load_async_to_lds_b128(
        (g_v4i_p)(size_t)(gamma + (size_t)t * BN + e0),
        (l_v4i_p)&sr[slot][tid], 0, 0);
  };

#pragma unroll
  for (int d = 0; d < DEPTH; ++d) issue(d, d);   // prime the ring

  for (int t = 0; t < T_STEPS; ++t) {
    if (t <= T_STEPS - DEPTH) {
      // Steady state: allow the 2*(DEPTH-1) copies (DEPTH-1 pairs) issued after step t's copy
      // to remain in flight; guarantees slot t%DEPTH has landed.
      __builtin_amdgcn_s_wait_asynccnt(2 * (DEPTH - 1));
    } else {
      __builtin_amdgcn_s_wait_asynccnt(0);   // tail: drain everything
    }
    asm volatile("" ::: "memory");           // keep ds reads below the wait
    const int slot = t & (DEPTH - 1);
    f4 g = sg[slot][tid];
    f4 r = sr[slot][tid];
    // Refill this slot (for step t+DEPTH) only after it has been consumed.
    if (t + DEPTH < T_STEPS) issue(t + DEPTH, slot);
    s = snn_step(m, s, g, r, w, c);
    __builtin_nontemporal_store(s, (f4*)(out + (size_t)t * BN + e0));
  }
#else
  // Fallback: software-pipelined direct global loads (register double buffer).
  const f4* gp = (const f4*)gate  + idx;
  const f4* rp = (const f4*)gamma + idx;
  const size_t st4 = BN / VEC;
  f4 g = gp[0];
  f4 r = rp[0];
  for (int t = 0; t < T_STEPS; ++t) {
    f4 gn = {}, rn = {};
    if (t + 1 < T_STEPS) {
      gn = gp[(size_t)(t + 1) * st4];
      rn = rp[(size_t)(t + 1) * st4];
    }
    s = snn_step(m, s, g, r, w, c);
    __builtin_nontemporal_store(s, (f4*)(out + (size_t)t * BN + e0));
    g = gn; r = rn;
  }
#endif
}

extern "C" void kernel_launch(void* const* d_in, const int* in_sizes, int n_in,
                              void* d_out, int out_size, void* d_ws, size_t ws_size,
                              hipStream_t stream) {
  // setup_inputs() order:
  //   0: membrane0        [B,N]   f32
  //   1: spikes0          [B,N]   f32
  //   2: sinusoidal_gate  [T,B,N] f32
  //   3: gamma            [T,B,N] f32
  //   4: membrane_weight  [N]     f32
  //   5: recurrent_weight [N]     f32
  const float* m0    = (const float*)d_in[0];
  const float* s0    = (const float*)d_in[1];
  const float* gate  = (const float*)d_in[2];
  const float* gamma = (const float*)d_in[3];
  const float* mw    = (const float*)d_in[4];
  const float* rw    = (const float*)d_in[5];
  float* out = (float*)d_out;  // [T,B,N] f32, fully overwritten

  snn_scan_kernel<<<NBLOCKS, BLOCK, 0, stream>>>(m0, s0, gate, gamma, mw, rw, out);
}